// ShapeAwareRoIAlign_35862976921814
// MI455X (gfx1250) — compile-verified
//
#include <hip/hip_runtime.h>
#include <hip/hip_bf16.h>

typedef __attribute__((ext_vector_type(2))) float v2f;
typedef __attribute__((ext_vector_type(8))) float v8f;

#define NBATCH 2
#define NCHAN  256
#define FH     200
#define FW     200
#define MH     128
#define MW     128
#define OUTH   7
#define OUTW   7
#define NBINS  49          // 7*7
#define NSAMP  196         // 49 bins * 4 subsamples
#define SCALE  0.25f

// One block per roi. 256 threads = 8 wave32.
// Phase 1: per-sample metadata (bilinear coeffs * mask weight * 0.25, corner offsets) -> LDS.
// Phase 2: WMMA_F32_16X16X4 pooling: D(16c x 16bins) += A(16c x 4 subs of bin g) * onehot_col(g).
__global__ __launch_bounds__(256) void shape_aware_roialign_kernel(
    const float* __restrict__ features,   // [2,256,200,200]
    const float* __restrict__ rois,       // [R,5]
    const float* __restrict__ masks,      // [R,128,128]
    float* __restrict__ out)              // [R,256,7,7]
{
    __shared__ float4 s_coef[NSAMP];
    __shared__ int4   s_offs[NSAMP];

    const int roi_id = blockIdx.x;
    const int t = threadIdx.x;

    const float* roi = rois + (size_t)roi_id * 5;
    const float rb  = roi[0];
    const float rx1 = roi[1], ry1 = roi[2], rx2 = roi[3], ry2 = roi[4];
    const int   b   = (int)rb;

    const float roi_start_w = rx1 * SCALE;
    const float roi_start_h = ry1 * SCALE;
    const float roi_w = fmaxf((rx2 - rx1 + 1.0f) * SCALE, 1.0f);
    const float roi_h = fmaxf((ry2 - ry1 + 1.0f) * SCALE, 1.0f);
    const float bin_w = roi_w * (1.0f / OUTW);
    const float bin_h = roi_h * (1.0f / OUTH);

    // ---------------- Phase 1: per-sample metadata ----------------
    if (t < NSAMP) {
        const int bin = t >> 2;          // 0..48, samples of a bin are consecutive
        const int sub = t & 3;
        const int ph = bin / OUTW, pw = bin - ph * OUTW;
        const int sy = sub >> 1,   sx = sub & 1;

        const float yy = roi_start_h + ((float)ph + ((float)sy + 0.5f) * 0.5f) * bin_h;
        const float xx = roi_start_w + ((float)pw + ((float)sx + 0.5f) * 0.5f) * bin_w;

        // feature-plane bilinear (zero outside (-1,H)x(-1,W))
        const bool valid = (yy > -1.0f) && (yy < (float)FH) && (xx > -1.0f) && (xx < (float)FW);
        const float y = fminf(fmaxf(yy, 0.0f), (float)(FH - 1));
        const float x = fminf(fmaxf(xx, 0.0f), (float)(FW - 1));
        const int y0 = (int)floorf(y);
        const int x0 = (int)floorf(x);
        const int y1 = min(y0 + 1, FH - 1);
        const int x1 = min(x0 + 1, FW - 1);
        const float ly = y - (float)y0, lx = x - (float)x0;
        const float hy = 1.0f - ly,     hx = 1.0f - lx;

        // mask weight at same sample point, roi-relative image coords
        float my = fminf(fmaxf(yy * 4.0f - ry1, 0.0f), ry2 - ry1);
        float mx = fminf(fmaxf(xx * 4.0f - rx1, 0.0f), rx2 - rx1);
        const bool mvalid = (my > -1.0f) && (my < (float)MH) && (mx > -1.0f) && (mx < (float)MW);
        my = fminf(fmaxf(my, 0.0f), (float)(MH - 1));
        mx = fminf(fmaxf(mx, 0.0f), (float)(MW - 1));
        const int my0 = (int)floorf(my);
        const int mx0 = (int)floorf(mx);
        const int my1 = min(my0 + 1, MH - 1);
        const int mx1 = min(mx0 + 1, MW - 1);
        const float mly = my - (float)my0, mlx = mx - (float)mx0;
        const float mhy = 1.0f - mly,      mhx = 1.0f - mlx;

        const float* mk = masks + (size_t)roi_id * (MH * MW);
        float wgt = mhy * mhx * mk[my0 * MW + mx0] + mhy * mlx * mk[my0 * MW + mx1]
                  + mly * mhx * mk[my1 * MW + mx0] + mly * mlx * mk[my1 * MW + mx1];
        if (!mvalid) wgt = 0.0f;

        const float q = valid ? 0.25f * wgt : 0.0f;   // 0.25 = 1/(S*S) bin mean folded in
        s_coef[t] = make_float4(q * hy * hx, q * hy * lx, q * ly * hx, q * ly * lx);
        s_offs[t] = make_int4(y0 * FW + x0, y0 * FW + x1, y1 * FW + x0, y1 * FW + x1);
    }
    __syncthreads();

    // ---------------- Phase 2: WMMA pooling ----------------
    const int w      = t >> 5;          // wave 0..7
    const int l      = t & 31;          // lane
    const int lane15 = l & 15;
    const bool hi    = (l >= 16);

    for (int p = 0; p < 8; ++p) {
        const int pair  = w * 8 + p;    // 64 pairs = 16 chan-chunks x 4 bin-tiles
        const int chunk = pair >> 2;    // 0..15 -> channels [16*chunk, 16*chunk+16)
        const int tile  = pair & 3;     // 0..3  -> bins     [16*tile,  16*tile+16)

        const int c = chunk * 16 + lane15;                 // A row M = lane&15
        const float* fb = features + ((size_t)b * NCHAN + c) * (FH * FW);

        v8f d = {};
        for (int gl = 0; gl < 16; ++gl) {
            const int g = tile * 16 + gl;                  // bin accumulated this step
            if (g >= NBINS) break;                         // wave-uniform: EXEC stays all-1s

            const int s0 = g * 4 + (hi ? 2 : 0);           // this lane-half's 2 subsamples
            const float4 c0 = s_coef[s0];
            const float4 c1 = s_coef[s0 + 1];
            const int4   o0 = s_offs[s0];
            const int4   o1 = s_offs[s0 + 1];

            const float v0 = c0.x * fb[o0.x] + c0.y * fb[o0.y]
                           + c0.z * fb[o0.z] + c0.w * fb[o0.w];
            const float v1 = c1.x * fb[o1.x] + c1.y * fb[o1.y]
                           + c1.z * fb[o1.z] + c1.w * fb[o1.w];

            v2f a;  a[0] = v0;  a[1] = v1;                 // A: 16x4 f32, row = channel
            const float sel = (lane15 == gl) ? 1.0f : 0.0f;
            v2f bb; bb[0] = sel; bb[1] = sel;              // B: one-hot column g (K-invariant)

            d = __builtin_amdgcn_wmma_f32_16x16x4_f32(
                    false, a, false, bb, (short)0, d, false, false);
        }

        const int col = tile * 16 + lane15;                // D col N = lane&15 = bin
        if (col < NBINS) {
            float* op = out + (((size_t)roi_id * NCHAN + chunk * 16 + (hi ? 8 : 0)) * NBINS) + col;
            #pragma unroll
            for (int v = 0; v < 8; ++v)                    // D row M = v + (hi?8:0)
                op[(size_t)v * NBINS] = d[v];
        }
    }
}

extern "C" void kernel_launch(void* const* d_in, const int* in_sizes, int n_in,
                              void* d_out, int out_size, void* d_ws, size_t ws_size,
                              hipStream_t stream) {
    const float* features = (const float*)d_in[0];
    const float* rois     = (const float*)d_in[1];
    const float* masks    = (const float*)d_in[2];
    float* out            = (float*)d_out;

    const int R = in_sizes[1] / 5;      // 512
    shape_aware_roialign_kernel<<<dim3(R), dim3(256), 0, stream>>>(
        features, rois, masks, out);
}